// ANIAngular_4063039062023
// MI455X (gfx1250) — compile-verified
//
#include <hip/hip_runtime.h>
#include <cstdint>
#include <cstddef>

typedef float v4f __attribute__((ext_vector_type(4)));
typedef int   v4i __attribute__((ext_vector_type(4)));

#define ASYNC_OK __has_builtin(__builtin_amdgcn_global_load_async_to_lds_b128)
#if ASYNC_OK
typedef __attribute__((address_space(1))) v4i gv4i;   // global (device) int4
typedef __attribute__((address_space(3))) v4i lv4i;   // LDS int4
#endif

constexpr float kEta    = 12.5f;
constexpr float kZeta   = 14.1f;
constexpr float kLog2e  = 1.44269504088896340736f;
constexpr float kInv2Pi = 0.15915494309189535f;   // 1/(2*pi)
constexpr int   kBlock  = 256;

// Output: (T, 32) f32, col = r*4 + s  (8 shifts major, 4 sections minor)
__global__ __launch_bounds__(kBlock)
void ani_angular_kernel(const float* __restrict__ dist,     // (2, T)
                        const float* __restrict__ vecs,     // (2, T, 3)
                        const float* __restrict__ shifts,   // (8,)
                        const float* __restrict__ sections, // (4,)
                        float* __restrict__ out,            // (T, 32)
                        int T)
{
    // 6 KB vector staging (256 triples * 2 legs * 12 B) + 16 broadcast consts
    __shared__ __align__(16) float smem[1536 + 16];
    float* svec   = smem;
    float* sconst = smem + 1536;

    const int       tid  = threadIdx.x;
    const long long base = (long long)blockIdx.x * kBlock;
    const long long t    = base + tid;
    const bool      full = (base + kBlock) <= (long long)T;

#if ASYNC_OK
    // --- CDNA5 async DMA: stage this block's tri_vectors into LDS ---------
    // leg0 bytes: [base*12, base*12+3072)   -> LDS [0, 3072)
    // leg1 bytes: [T*12 + base*12, +3072)   -> LDS [3072, 6144)
    if (full) {
        const char*     gv = (const char*)vecs;
        const long long b0 = base * 12LL;
        const long long b1 = (long long)T * 12LL + base * 12LL;
        char*           lb = (char*)smem;
        {
            const int       c   = tid;                     // chunks 0..255
            const long long src = (c < 192) ? (b0 + (long long)c * 16)
                                            : (b1 + (long long)(c - 192) * 16);
            __builtin_amdgcn_global_load_async_to_lds_b128(
                (gv4i*)(gv + src), (lv4i*)(lb + c * 16), 0, 0);
        }
        if (tid < 128) {
            const int       c   = 256 + tid;               // chunks 256..383 (all leg1)
            const long long src = b1 + (long long)(c - 192) * 16;
            __builtin_amdgcn_global_load_async_to_lds_b128(
                (gv4i*)(gv + src), (lv4i*)(lb + c * 16), 0, 0);
        }
    }
#endif

    // --- broadcast constants: cos/sin of sections, shifts ------------------
    if (tid < 4) {
        float th = sections[tid] * kInv2Pi;                // revolutions
        sconst[tid]     = __builtin_amdgcn_cosf(th);
        sconst[4 + tid] = __builtin_amdgcn_sinf(th);
    } else if (tid >= 8 && tid < 16) {
        sconst[tid] = shifts[tid - 8];                     // sconst[8..15]
    }

#if ASYNC_OK
  #if __has_builtin(__builtin_amdgcn_s_wait_asynccnt)
    __builtin_amdgcn_s_wait_asynccnt(0);
  #elif defined(__gfx1250__)
    asm volatile("s_wait_asynccnt 0" ::: "memory");
  #endif
#endif
    __syncthreads();

    if (t >= T) return;

    const v4f csv = ((const v4f*)sconst)[0];   // cos(theta_s)
    const v4f ssv = ((const v4f*)sconst)[1];   // sin(theta_s)
    const v4f sh0 = ((const v4f*)sconst)[2];   // shifts 0..3
    const v4f sh1 = ((const v4f*)sconst)[3];   // shifts 4..7

    float v0x, v0y, v0z, v1x, v1y, v1z;
#if ASYNC_OK
    if (full) {
        const int o0 = tid * 3, o1 = 768 + tid * 3;        // stride-3 dwords: bank-conflict-free
        v0x = svec[o0]; v0y = svec[o0 + 1]; v0z = svec[o0 + 2];
        v1x = svec[o1]; v1y = svec[o1 + 1]; v1z = svec[o1 + 2];
    } else
#endif
    {
        const size_t p0 = (size_t)t * 3;
        const size_t p1 = (size_t)T * 3 + (size_t)t * 3;
        v0x = vecs[p0]; v0y = vecs[p0 + 1]; v0z = vecs[p0 + 2];
        v1x = vecs[p1]; v1y = vecs[p1 + 1]; v1z = vecs[p1 + 2];
    }

    const float d0 = __builtin_nontemporal_load(dist + t);
    const float d1 = __builtin_nontemporal_load(dist + T + t);

    // fcut = 0.5*cos(d*pi/3.5)+0.5 ; v_cos takes revolutions -> arg = d/7
    const float fc0  = fmaf(__builtin_amdgcn_cosf(d0 * (1.0f / 7.0f)), 0.5f, 0.5f);
    const float fc1  = fmaf(__builtin_amdgcn_cosf(d1 * (1.0f / 7.0f)), 0.5f, 0.5f);
    const float pref = 2.0f * fc0 * fc1;                   // folds the leading 2.0

    // c = cos(angle) = 0.95 * dot/(d0*d1) ; sin(angle) = sqrt(1-c^2) (angle in [0,pi])
    const float dot = fmaf(v0x, v1x, fmaf(v0y, v1y, v0z * v1z));
    const float dd  = fmaxf(d0 * d1, 1e-10f);
    const float c   = 0.95f * dot * __builtin_amdgcn_rcpf(dd);
    const float sn  = __builtin_amdgcn_sqrtf(fmaxf(fmaf(-c, c, 1.0f), 0.0f));

    // angular[s] = pref * ((1 + cos(angle - theta_s))/2)^zeta
    float ang[4];
#pragma unroll
    for (int s = 0; s < 4; ++s) {
        const float cosd = fmaf(c, csv[s], sn * ssv[s]);   // cos(angle - theta_s)
        const float hc   = fmaxf(fmaf(cosd, 0.5f, 0.5f), 0.0f);
        ang[s] = pref * __builtin_amdgcn_exp2f(kZeta * __builtin_amdgcn_logf(hc));
    }

    // radial[r] = exp(-eta*(mean_d - shift_r)^2) ; stream out 8 x b128 NT stores
    const float m  = 0.5f * (d0 + d1);
    float* op = out + (size_t)t * 32;
#pragma unroll
    for (int r = 0; r < 8; ++r) {
        const float s_r = (r < 4) ? sh0[r] : sh1[r - 4];
        const float u   = m - s_r;
        const float rad = __builtin_amdgcn_exp2f((-kEta * kLog2e) * (u * u));
        v4f o = { rad * ang[0], rad * ang[1], rad * ang[2], rad * ang[3] };
        __builtin_nontemporal_store(o, (v4f*)(op + r * 4));
    }
}

extern "C" void kernel_launch(void* const* d_in, const int* in_sizes, int n_in,
                              void* d_out, int out_size, void* d_ws, size_t ws_size,
                              hipStream_t stream)
{
    const float* dist     = (const float*)d_in[0];   // (2,T)
    const float* vecs     = (const float*)d_in[1];   // (2,T,3)
    const float* shifts   = (const float*)d_in[2];   // (8,)
    const float* sections = (const float*)d_in[3];   // (4,)
    float*       out      = (float*)d_out;           // (T,32)

    const int T      = in_sizes[0] / 2;
    const int blocks = (T + kBlock - 1) / kBlock;    // 15625 for T=4M

    ani_angular_kernel<<<blocks, kBlock, 0, stream>>>(dist, vecs, shifts, sections, out, T);
}